// CBOW_87351044866530
// MI455X (gfx1250) — compile-verified
//
#include <hip/hip_runtime.h>

typedef __attribute__((ext_vector_type(2))) float v2f;
typedef __attribute__((ext_vector_type(8))) float v8f;

#define VOCAB 32000
#define EMB   128
#define BATCH 2048

// ---------------------------------------------------------------------------
// Kernel A: per batch element, recover the two one-hot indices from x and
// build avg[b, 0:128] = proj[i0] + proj[i1]  (WINDOW_SIZE == 1, so no divide).
// Reads x exactly once (the 524 MB term of the roofline), coalesced float4.
// ---------------------------------------------------------------------------
__global__ void cbow_avg_kernel(const float* __restrict__ x,
                                const float* __restrict__ proj,
                                float* __restrict__ avg) {
  const int b = blockIdx.x;
  __shared__ int s_idx[2];
  if (threadIdx.x < 2) s_idx[threadIdx.x] = 0;
  __syncthreads();

  for (int c = 0; c < 2; ++c) {
    const float4* row = (const float4*)(x + ((size_t)b * 2 + c) * VOCAB);
    for (int i = threadIdx.x; i < VOCAB / 4; i += blockDim.x) {
      float4 v = row[i];
      // exactly one element of the row is nonzero -> single writer, no race
      if (v.x != 0.f) s_idx[c] = 4 * i + 0;
      if (v.y != 0.f) s_idx[c] = 4 * i + 1;
      if (v.z != 0.f) s_idx[c] = 4 * i + 2;
      if (v.w != 0.f) s_idx[c] = 4 * i + 3;
    }
  }
  __syncthreads();

  const int i0 = s_idx[0];
  const int i1 = s_idx[1];
  for (int e = threadIdx.x; e < EMB; e += blockDim.x) {
    avg[(size_t)b * EMB + e] =
        proj[(size_t)i0 * EMB + e] + proj[(size_t)i1 * EMB + e];
  }
}

// ---------------------------------------------------------------------------
// Kernel B: out[2048, 32000] = avg[2048,128] @ W^T[128,32000] + bias,
// exact fp32 via V_WMMA_F32_16X16X4_F32 (wave32, D = A*B + C).
//
// Fragment layouts (CDNA5 ISA §7.12.2, 32-bit data):
//   A 16x4 : lane(0..15)=M, VGPR0={K0|K2 per lane-half}, VGPR1={K1|K3}
//   B 4x16 : lane(0..15)=N, VGPR0={K0|K2 per lane-half}, VGPR1={K1|K3}
//   C 16x16: VGPR r -> M=r (lanes 0..15) / M=r+8 (lanes 16..31), N=lane%16
//
// Each wave computes a 16x64 strip: A fragment (full K=128) held in 32 v2f
// registers, four independent accumulator chains over the N subtiles.
// ---------------------------------------------------------------------------
__global__ void cbow_gemm_wmma_kernel(const float* __restrict__ avg,
                                      const float* __restrict__ W,
                                      const float* __restrict__ bias,
                                      float* __restrict__ out) {
  const int waveId = blockIdx.x * (blockDim.x >> 5) + (threadIdx.x >> 5);
  const int lane = threadIdx.x & 31;
  const int half = lane >> 4;   // selects K pair {0,1} vs {2,3} within a step
  const int l    = lane & 15;   // M index for A, N index for B/C

  const int mTile  = waveId / (VOCAB / 64);  // 0..127
  const int nSuper = waveId % (VOCAB / 64);  // 0..499
  const int mbase  = mTile * 16;
  const int nbase  = nSuper * 64;

  // --- preload full-K A fragment: a[kk] covers K = 4*kk .. 4*kk+3 ---
  const v2f* aptr = (const v2f*)(avg + (size_t)(mbase + l) * EMB) + half;
  v2f a[32];
#pragma unroll
  for (int kk = 0; kk < 32; ++kk) a[kk] = aptr[kk * 2];

  // --- accumulators initialized with bias (bias is per-column N) ---
  v8f acc[4];
#pragma unroll
  for (int s = 0; s < 4; ++s) {
    const float bv = bias[nbase + s * 16 + l];
#pragma unroll
    for (int r = 0; r < 8; ++r) acc[s][r] = bv;
  }

  // B column N=v is W row v (out[b,v] = sum_e avg[b,e] * W[v,e])
  const v2f* b0 = (const v2f*)(W + (size_t)(nbase +  0 + l) * EMB) + half;
  const v2f* b1 = (const v2f*)(W + (size_t)(nbase + 16 + l) * EMB) + half;
  const v2f* b2 = (const v2f*)(W + (size_t)(nbase + 32 + l) * EMB) + half;
  const v2f* b3 = (const v2f*)(W + (size_t)(nbase + 48 + l) * EMB) + half;

#pragma unroll
  for (int kk = 0; kk < 32; ++kk) {
    v2f f0 = b0[kk * 2];
    v2f f1 = b1[kk * 2];
    v2f f2 = b2[kk * 2];
    v2f f3 = b3[kk * 2];
    // 8-arg form: (neg_a, A, neg_b, B, c_mod, C, reuse_a, reuse_b)
    acc[0] = __builtin_amdgcn_wmma_f32_16x16x4_f32(
        false, a[kk], false, f0, (short)0, acc[0], false, false);
    acc[1] = __builtin_amdgcn_wmma_f32_16x16x4_f32(
        false, a[kk], false, f1, (short)0, acc[1], false, false);
    acc[2] = __builtin_amdgcn_wmma_f32_16x16x4_f32(
        false, a[kk], false, f2, (short)0, acc[2], false, false);
    acc[3] = __builtin_amdgcn_wmma_f32_16x16x4_f32(
        false, a[kk], false, f3, (short)0, acc[3], false, false);
  }

  // --- store C per the 16x16 f32 C/D layout ---
#pragma unroll
  for (int s = 0; s < 4; ++s) {
    const int col = nbase + s * 16 + l;
#pragma unroll
    for (int r = 0; r < 8; ++r) {
      const int row = mbase + r + 8 * half;
      out[(size_t)row * VOCAB + col] = acc[s][r];
    }
  }
}

// ---------------------------------------------------------------------------
extern "C" void kernel_launch(void* const* d_in, const int* in_sizes, int n_in,
                              void* d_out, int out_size, void* d_ws, size_t ws_size,
                              hipStream_t stream) {
  const float* x    = (const float*)d_in[0];  // [2048, 2, 32000]
  const float* proj = (const float*)d_in[1];  // [32000, 128]
  const float* W    = (const float*)d_in[2];  // [32000, 128]
  const float* bias = (const float*)d_in[3];  // [32000]
  float* out = (float*)d_out;                 // [2048, 32000]
  float* avg = (float*)d_ws;                  // [2048, 128] scratch (1 MB)

  cbow_avg_kernel<<<BATCH, 256, 0, stream>>>(x, proj, avg);

  // 2048/16 mTiles * 32000/64 nSupers = 64000 waves = 8000 blocks of 8 waves
  const int totalWaves = (BATCH / 16) * (VOCAB / 64);
  cbow_gemm_wmma_kernel<<<totalWaves / 8, 256, 0, stream>>>(avg, W, bias, out);
}